// SmoothL1LossWordAndSentenceAndImage_9861244912212
// MI455X (gfx1250) — compile-verified
//
#include <hip/hip_runtime.h>
#include <hip/hip_bf16.h>

typedef __attribute__((ext_vector_type(2))) float v2f;
typedef __attribute__((ext_vector_type(8))) float v8f;

#define BB 16
#define TT 2048
#define DD 512
#define TC 64   // timesteps per block (T-chunk)

// ws layout (floats): [0]=word_sum, [1 .. 1+B*D)=pred_sum, [1+B*D .. 1+2*B*D)=tgt_sum
#define WS_WORD 0
#define WS_PRED 1
#define WS_TGT  (1 + BB * DD)
#define WS_TOTAL (1 + 2 * BB * DD)

__device__ __forceinline__ float sl1(float x, float y) {
  float d = fabsf(x - y);
  return (d < 1.0f) ? 0.5f * d * d : d - 0.5f;
}

__global__ void zero_ws_kernel(float* __restrict__ ws, int n) {
  int i = blockIdx.x * blockDim.x + threadIdx.x;
  if (i < n) ws[i] = 0.0f;
}

// Streaming pass: column sums over T (per (b,d)) + masked SmoothL1 word sum.
// blockDim = 256 (8 waves, wave32). grid = (T/TC, B).
__global__ void pass1_kernel(const float* __restrict__ preds,
                             const float* __restrict__ tgts,
                             const int* __restrict__ dl,
                             float* __restrict__ ws) {
  const int b    = blockIdx.y;
  const int t0   = blockIdx.x * TC;
  const int len  = dl[b];                 // block-uniform
  const int tend = min(t0 + TC, len);     // block-uniform clip -> scalar branch only
  const int tid  = threadIdx.x;
  const int d    = tid * 2;               // two consecutive columns per thread

  float sp0 = 0.f, sp1 = 0.f, st0 = 0.f, st1 = 0.f, w = 0.f;
  if (tend > t0) {
    size_t base = ((size_t)b * TT + (size_t)t0) * DD + (size_t)d;
    const float* p = preds + base;
    const float* g = tgts + base;
    for (int t = t0; t < tend; ++t) {
      float p0 = p[0], p1 = p[1];         // global_load_b64, wave-coalesced 256B
      float g0 = g[0], g1 = g[1];
      sp0 += p0; sp1 += p1;
      st0 += g0; st1 += g1;
      w += sl1(p0, g0) + sl1(p1, g1);
      p += DD; g += DD;
    }
  }

  // Each thread owns distinct (b,d) columns: accumulate directly.
  float* pred_sum = ws + WS_PRED + (size_t)b * DD;
  float* tgt_sum  = ws + WS_TGT  + (size_t)b * DD;
  atomicAdd(&pred_sum[d],     sp0);
  atomicAdd(&pred_sum[d + 1], sp1);
  atomicAdd(&tgt_sum[d],      st0);
  atomicAdd(&tgt_sum[d + 1],  st1);

  // Wave-level fold of the word-loss partial with native f32 WMMA.
  // A layout (16x4 f32): lane L (half=L>>4, m=L&15): elem j -> A[m, 2*half+j].
  // Set A[:,{0,2}] = lane partials, B = ones  =>  D[m,n] = w_m + w_{m+16}.
  v2f a;    a.x = w;    a.y = 0.0f;
  v2f ones; ones.x = 1.0f; ones.y = 1.0f;
  v8f c = {};
  c = __builtin_amdgcn_wmma_f32_16x16x4_f32(false, a, false, ones,
                                            (short)0, c, false, false);
  // Row-fold the 8 C VGPRs, then a second WMMA folds the two half-waves.
  float s = c[0] + c[1] + c[2] + c[3] + c[4] + c[5] + c[6] + c[7];
  v2f a2; a2.x = s; a2.y = 0.0f;
  v8f c2 = {};
  c2 = __builtin_amdgcn_wmma_f32_16x16x4_f32(false, a2, false, ones,
                                             (short)0, c2, false, false);
  float wave_sum = c2[0];                 // full 32-lane sum, identical in all lanes

  __shared__ float s_wave[8];
  const int lane = tid & 31;
  const int wave = tid >> 5;
  if (lane == 0) s_wave[wave] = wave_sum;
  __syncthreads();
  if (tid == 0) {
    float tot = 0.f;
    for (int i = 0; i < 8; ++i) tot += s_wave[i];
    atomicAdd(&ws[WS_WORD], tot);
  }
}

// Tiny finalize: img norms, ragged means, three SmoothL1 means, final scalar.
__global__ void finalize_kernel(const float* __restrict__ img,
                                const int* __restrict__ dl,
                                const float* __restrict__ ws,
                                float* __restrict__ out) {
  __shared__ float s_norm[BB];
  __shared__ float s_acc;
  const int tid = threadIdx.x;            // 512 threads
  if (tid < BB) s_norm[tid] = 0.0f;
  if (tid == 0) s_acc = 0.0f;
  __syncthreads();

  for (int b = 0; b < BB; ++b) {
    float x = img[b * DD + tid];
    atomicAdd(&s_norm[b], x * x);
  }
  __syncthreads();
  if (tid < BB) s_norm[tid] = 1.0f / fmaxf(sqrtf(s_norm[tid]), 1e-12f);
  __syncthreads();

  float acc = 0.0f;
  for (int i = tid; i < BB * DD; i += 512) {
    int b = i >> 9;                       // i / DD
    float len = (float)dl[b];
    float pm = ws[WS_PRED + i] / len;
    float tm = ws[WS_TGT + i] / len;
    float in = img[i] * s_norm[b];
    acc += sl1(pm, tm) + sl1(pm, in) + sl1(tm, in);
  }
  atomicAdd(&s_acc, acc);
  __syncthreads();

  if (tid == 0) {
    float lens = 0.0f;
    for (int b = 0; b < BB; ++b) lens += (float)dl[b];
    float word = ws[WS_WORD] / (lens * (float)DD);
    out[0] = word + s_acc / (float)(BB * DD);
  }
}

extern "C" void kernel_launch(void* const* d_in, const int* in_sizes, int n_in,
                              void* d_out, int out_size, void* d_ws, size_t ws_size,
                              hipStream_t stream) {
  const float* preds = (const float*)d_in[0];
  const float* tgts  = (const float*)d_in[1];
  const float* img   = (const float*)d_in[2];
  const int*   dl    = (const int*)d_in[3];
  float* ws  = (float*)d_ws;
  float* out = (float*)d_out;

  zero_ws_kernel<<<(WS_TOTAL + 255) / 256, 256, 0, stream>>>(ws, WS_TOTAL);

  dim3 grid(TT / TC, BB);                 // 32 x 16 = 512 blocks, 4096 waves
  pass1_kernel<<<grid, 256, 0, stream>>>(preds, tgts, dl, ws);

  finalize_kernel<<<1, 512, 0, stream>>>(img, dl, ws, out);
}